// PointUpsample_6176162972236
// MI455X (gfx1250) — compile-verified
//
#include <hip/hip_runtime.h>

typedef float v2f __attribute__((ext_vector_type(2)));
typedef float v8f __attribute__((ext_vector_type(8)));

namespace {
constexpr int   BS   = 2;
constexpr int   M    = 4096;    // source points
constexpr int   N    = 16384;   // parent points
constexpr int   C    = 64;      // channels
constexpr int   WAVES = 8;      // waves per workgroup (wave32)
constexpr int   TPB  = WAVES * 32;   // 256 threads
constexpr int   NPB  = WAVES * 16;   // parent points per workgroup
constexpr int   NTILE = M / 16;      // 256 source tiles
constexpr float KEPS = 1e-8f;
}

// Branchless sorted insert into a 3-element (key, idx) min-list.
__device__ __forceinline__ void insert3(float rv, int ridx,
                                        float& d1, float& d2, float& d3,
                                        int& i1, int& i2, int& i3)
{
    bool c1 = rv < d1, c2 = rv < d2, c3 = rv < d3;
    float nd3 = c2 ? d2 : (c3 ? rv : d3); int ni3 = c2 ? i2 : (c3 ? ridx : i3);
    float nd2 = c1 ? d1 : (c2 ? rv : d2); int ni2 = c1 ? i1 : (c2 ? ridx : i2);
    float nd1 = c1 ? rv : d1;             int ni1 = c1 ? ridx : i1;
    d3 = nd3; i3 = ni3;
    d2 = nd2; i2 = ni2;
    d1 = nd1; i1 = ni1;
}

__global__ __launch_bounds__(TPB)
void point_upsample_kernel(const float* __restrict__ xyz,        // (BS, M, 3)
                           const float* __restrict__ pxyz,       // (BS, N, 3)
                           const float* __restrict__ feats,      // (BS, C, M)
                           float* __restrict__ out)              // (BS, C, N)
{
    // Arranged A-matrix table: per source point [-2x, -2y, -2z, |x|^2]  (64 KB)
    __shared__ float atab[M * 4];

    const int tid  = threadIdx.x;
    const int lane = tid & 31;
    const int wave = tid >> 5;
    const int blkPerBatch = N / NPB;                 // 128
    const int b    = blockIdx.x / blkPerBatch;
    const int nblk = (blockIdx.x % blkPerBatch) * NPB;

    // ---- cooperative LDS A-table build ----
    const float* xb = xyz + (size_t)b * M * 3;
    for (int i = tid; i < M; i += TPB) {
        float x = xb[i * 3 + 0];
        float y = xb[i * 3 + 1];
        float z = xb[i * 3 + 2];
        float4 t;
        t.x = -2.0f * x;
        t.y = -2.0f * y;
        t.z = -2.0f * z;
        t.w = x * x + y * y + z * z;
        *(float4*)&atab[i * 4] = t;
    }
    __syncthreads();

    // ---- per-wave setup: this wave owns 16 parent columns ----
    const int  nbase = nblk + wave * 16;
    const int  ncol  = nbase + (lane & 15);
    const bool hi    = lane >= 16;

    const float* pp = pxyz + (size_t)b * N * 3 + (size_t)ncol * 3;
    float px = pp[0], py = pp[1], pz = pp[2];
    float pn2 = px * px + py * py + pz * pz;

    // B fragment (4x16, K split across lane-halves): lanes 0-15 -> K=0,1 ; 16-31 -> K=2,3
    v2f bf;
    bf.x = hi ? pz   : px;
    bf.y = hi ? 1.0f : py;

    // running top-3 on ranking key r = |x|^2 - 2 x.p  (add |p|^2 after the scan)
    float d1 = 1e30f, d2 = 1e30f, d3 = 1e30f;
    int   i1 = 0, i2 = 0, i3 = 0;

    const int aoff   = (lane & 15) * 4 + (hi ? 2 : 0);  // floats into a 16-pt tile
    const int rowoff = hi ? 8 : 0;                      // D rows this lane holds

    auto loadA = [&](int t) -> v2f {
        return *(const v2f*)&atab[t * 64 + aoff];
    };

    // WMMA + two-level guarded top-3 scan for one 16-source-point tile.
    auto process = [&](v2f af, int tile) {
        v8f acc = {};
        // D[m][n] = |x_m|^2 - 2 * dot(x_m, p_n)
        acc = __builtin_amdgcn_wmma_f32_16x16x4_f32(
                  false, af, false, bf, (short)0, acc, false, false);

        float vmin = fminf(fminf(fminf(acc[0], acc[1]), fminf(acc[2], acc[3])),
                           fminf(fminf(acc[4], acc[5]), fminf(acc[6], acc[7])));
        if (__any(vmin < d3)) {
            const int mb = tile * 16 + rowoff;
            #pragma unroll
            for (int r = 0; r < 8; ++r) {
                float rv = acc[r];
                if (__any(rv < d3)) {
                    insert3(rv, mb + r, d1, d2, d3, i1, i2, i3);
                }
            }
        }
    };

    // ---- software-pipelined scan over all source tiles (prefetch depth 2) ----
    v2f af0 = loadA(0);
    v2f af1 = loadA(1);
    for (int t = 0; t < NTILE; t += 2) {
        const int tp = (t + 2 < NTILE) ? (t + 2) : 0;  // dummy wrap on last iter
        v2f nf0 = loadA(tp);
        v2f nf1 = loadA(tp + 1);
        process(af0, t);
        process(af1, t + 1);
        af0 = nf0;
        af1 = nf1;
    }

    // ---- merge lane pair (l, l^16): same column, disjoint row halves ----
    float o1 = __shfl_xor(d1, 16), o2 = __shfl_xor(d2, 16), o3 = __shfl_xor(d3, 16);
    int   oi1 = __shfl_xor(i1, 16), oi2 = __shfl_xor(i2, 16), oi3 = __shfl_xor(i3, 16);
    insert3(o1, oi1, d1, d2, d3, i1, i2, i3);
    insert3(o2, oi2, d1, d2, d3, i1, i2, i3);
    insert3(o3, oi3, d1, d2, d3, i1, i2, i3);

    // ---- weights (match reference: dist = sqrt(max(d2,0)); inv-dist normalized) ----
    float dd1 = sqrtf(fmaxf(d1 + pn2, 0.0f));
    float dd2 = sqrtf(fmaxf(d2 + pn2, 0.0f));
    float dd3 = sqrtf(fmaxf(d3 + pn2, 0.0f));
    float w1 = 1.0f / (dd1 + KEPS);
    float w2 = 1.0f / (dd2 + KEPS);
    float w3 = 1.0f / (dd3 + KEPS);
    float ws = 1.0f / (w1 + w2 + w3);
    w1 *= ws; w2 *= ws; w3 *= ws;

    // ---- feature gather + blend: lower lanes ch 0-31, upper lanes ch 32-63 ----
    const float* fb = feats + (size_t)b * C * M;
    float*       ob = out   + (size_t)b * C * N + ncol;
    const int ch0 = hi ? (C / 2) : 0;

    #pragma unroll 4
    for (int ch = 0; ch < C / 2; ++ch) {
        const float* fr = fb + (size_t)(ch0 + ch) * M;
        float v = w1 * fr[i1] + w2 * fr[i2] + w3 * fr[i3];
        ob[(size_t)(ch0 + ch) * N] = v;
    }
}

extern "C" void kernel_launch(void* const* d_in, const int* in_sizes, int n_in,
                              void* d_out, int out_size, void* d_ws, size_t ws_size,
                              hipStream_t stream)
{
    (void)in_sizes; (void)n_in; (void)out_size; (void)d_ws; (void)ws_size;
    const float* xyz   = (const float*)d_in[0];
    const float* pxyz  = (const float*)d_in[1];
    const float* feats = (const float*)d_in[2];
    float*       out   = (float*)d_out;

    dim3 grid(BS * (N / NPB));   // 2 * 128 = 256 workgroups
    dim3 block(TPB);             // 256 threads = 8 wave32
    hipLaunchKernelGGL(point_upsample_kernel, grid, block, 0, stream,
                       xyz, pxyz, feats, out);
}